// iSHT_75660143886824
// MI455X (gfx1250) — compile-verified
//
#include <hip/hip_runtime.h>
#include <hip/hip_bf16.h>

#define NSIDE 128
#define LMAXD 511              // lmax == ntheta (number of rings) == l-count
#define MMAXD 257              // mmax
#define NPIXD (12 * NSIDE * NSIDE)

typedef float v2f __attribute__((ext_vector_type(2)));
typedef float v8f __attribute__((ext_vector_type(8)));

// ---------------------------------------------------------------------------
// Stage 1: inverse Legendre transform via V_WMMA_F32_16X16X4_F32.
//   ftm[k,m].{re,im} = sum_l x_{re,im}[l,m] * pct[m,l,k]
// Per-wave tile: D(16x16) = A(16x4) x B(4x16), K-chunks of 4 over l.
//   A[r][l'] = pct[m, l0+l', k0+r]   (M = ring k, K = l)
//   B[l'][0] = x_re[l0+l'], B[l'][1] = x_im[l0+l'], N>=2 columns are 0.
// B columns are selected purely by a per-lane LDS base pointer
// (re / im / zero array), so the hot loop has no divergence at all:
//   s_clause{8x global_load_b32} + 4x ds_load_b64 + 4x wmma per iteration.
// Memory-bound on pct (268 MB read once -> ~11.5 us at 23.3 TB/s).
// ---------------------------------------------------------------------------
__global__ __launch_bounds__(256) void legendre_wmma_kernel(
    const float* __restrict__ x_re, const float* __restrict__ x_im,
    const float* __restrict__ pct, float* __restrict__ ftm)
{
    __shared__ float sx[3 * 512];                   // [re|im|zeros], idx 511 = 0

    const int m = blockIdx.x;                       // 0..256
    for (int i = threadIdx.x; i < 512; i += 256) {
        sx[i]        = (i < LMAXD) ? x_re[i * MMAXD + m] : 0.f;
        sx[512 + i]  = (i < LMAXD) ? x_im[i * MMAXD + m] : 0.f;
        sx[1024 + i] = 0.f;
    }
    __syncthreads();

    const int wave = threadIdx.x >> 5;              // 0..7 (wave32)
    const int lane = threadIdx.x & 31;
    const int hlf  = lane >> 4;                     // K half-select
    const int r    = lane & 15;                     // M row / N column
    const int k0   = (wave + 8 * blockIdx.y) * 16;  // k-tile base, 0..496
    int kk = k0 + r;
    if (kk > LMAXD - 1) kk = LMAXD - 1;             // clamp; row k=511 never stored

    // Per-lane B source: N==0 -> re, N==1 -> im, N>=2 -> zeros.
    const float* xb = &sx[((r < 2) ? r : 2) * 512];

    const long mbase = (long)m * LMAXD * LMAXD;
    const float* pA = pct + mbase + (long)(2 * hlf) * LMAXD + kk;

    v8f c0 = {0.f,0.f,0.f,0.f,0.f,0.f,0.f,0.f};
    v8f c1 = {0.f,0.f,0.f,0.f,0.f,0.f,0.f,0.f};
    int row = 2 * hlf;                              // current K=l row (always even)

    // One K=4 chunk: 2 A loads (rows ROFF, ROFF+1) + 1 aligned b64 B load.
#define CHUNK(ROFF, CACC)                                                     \
    do {                                                                      \
        v2f a, b;                                                             \
        a.x = pA[(ROFF) * LMAXD];                                             \
        a.y = pA[((ROFF) + 1) * LMAXD];                                       \
        const float2 vb = *(const float2*)(xb + row + (ROFF));                \
        b.x = vb.x; b.y = vb.y;                                               \
        CACC = __builtin_amdgcn_wmma_f32_16x16x4_f32(                         \
                   false, a, false, b, (short)0, CACC, false, false);         \
    } while (0)

    // Main loop: 4 chunks (16 l-rows) per iteration.
    // 28 iterations with prefetch (+64 rows stays inside this m-block)...
    #pragma unroll 1
    for (int i = 0; i < 28; ++i) {
        __builtin_prefetch(pA + 64 * LMAXD, 0, 3);  // global_prefetch_b8
        CHUNK(0, c0); CHUNK(4, c1); CHUNK(8, c0); CHUNK(12, c1);
        pA += 16 * LMAXD; row += 16;
    }
    // ...then 3 iterations without prefetch (chunks 112..123, rows 448..495).
    #pragma unroll 1
    for (int i = 0; i < 3; ++i) {
        CHUNK(0, c0); CHUNK(4, c1); CHUNK(8, c0); CHUNK(12, c1);
        pA += 16 * LMAXD; row += 16;
    }
    // Epilogue: chunks 124..126 (rows 496..507), fully in-bounds.
    CHUNK(0, c0); CHUNK(4, c1); CHUNK(8, c0);
    pA += 12 * LMAXD; row += 12;
    {   // Tail chunk 127: rows 508..511. Phantom row 511: B reads the zero
        // sentinel; A re-reads row 510 via pointer select (no OOB access).
        v2f a, b;
        a.x = pA[0];
        const float* pAy = pA + (hlf ? 0 : LMAXD);
        a.y = pAy[0];
        const float2 vb = *(const float2*)(xb + row);   // xb[511] == 0
        b.x = vb.x; b.y = vb.y;
        c1 = __builtin_amdgcn_wmma_f32_16x16x4_f32(
                 false, a, false, b, (short)0, c1, false, false);
    }
#undef CHUNK

    // D layout: lanes 0-15 -> N=lane, M=q; lanes 16-31 -> N=lane-16, M=8+q.
    if (r < 2) {
#pragma unroll
        for (int q = 0; q < 8; ++q) {
            const int k = k0 + q + 8 * hlf;
            if (k < LMAXD)
                ftm[((long)k * MMAXD + m) * 2 + r] = c0[q] + c1[q];
        }
    }
}

// ---------------------------------------------------------------------------
// Stage 2: per-ring real DFT synthesis.
// The reference's Bluestein chirp-z is algebraically a length-nphi DFT, and
// hermitian symmetry of the extended spectrum reduces the output to
//   map[cum+p] = sum_{n=0}^{nphi/2} w_n*(a_n*cos(2pi*n*p/nphi)
//                                      - b_n*sin(2pi*n*p/nphi)),
// (a,b) = ftm[t,n]*exp(i*n*phi0_t), w = 1 for n==0 or n==nphi/2 else 2.
// One block per ring; blocked rotation recurrence, exact resync per 64 terms.
// ---------------------------------------------------------------------------
__global__ __launch_bounds__(256) void ring_synth_kernel(
    const float* __restrict__ ftm, float* __restrict__ out)
{
    const int t = blockIdx.x;                       // ring 0..510
    const float PI = 3.14159265358979323846f;

    int nphi, cum; float phi0;
    const int n1 = NSIDE - 1;
    if (t < n1) {                                   // north polar cap
        nphi = 4 * (t + 1);
        cum  = 2 * t * (t + 1);
        phi0 = PI / (4.0f * (float)(t + 1));
    } else if (t <= 3 * NSIDE - 1) {                // equatorial belt
        nphi = 4 * NSIDE;
        cum  = 2 * n1 * NSIDE + (t - n1) * 4 * NSIDE;
        phi0 = ((t - NSIDE + 2) & 1) ? PI / (4.0f * (float)NSIDE) : 0.0f;
    } else {                                        // south polar cap
        const int q = 4 * NSIDE - 1 - t;
        nphi = 4 * q;
        cum  = NPIXD - 2 * q * (q + 1);
        phi0 = PI / (4.0f * (float)q);
    }
    const int halfn = nphi / 2 + 1;                 // <= 257 == MMAXD

    __shared__ float sa[MMAXD + 3];
    __shared__ float sb[MMAXD + 3];

    for (int n = threadIdx.x; n < halfn; n += 256) {
        const float re = ftm[((long)t * MMAXD + n) * 2 + 0];
        const float im = ftm[((long)t * MMAXD + n) * 2 + 1];
        float sph, cph;
        __sincosf((float)n * phi0, &sph, &cph);     // ring phase exp(i*n*phi0)
        const float w = (n == 0 || 2 * n == nphi) ? 1.0f : 2.0f;
        sa[n] = w * (re * cph - im * sph);
        sb[n] = w * (re * sph + im * cph);
    }
    __syncthreads();

    const float twopi = 6.28318530717958647692f;
    const float invN  = 1.0f / (float)nphi;

    for (int p = threadIdx.x; p < nphi; p += 256) {
        float sd, cd;
        __sincosf(twopi * (float)p * invN, &sd, &cd);   // per-step rotation
        float sum = 0.0f;
        for (int base = 0; base < halfn; base += 64) {
            const int rm = (base * p) % nphi;           // exact angle resync
            float sn, cs;
            __sincosf(twopi * (float)rm * invN, &sn, &cs);
            int cnt = halfn - base; if (cnt > 64) cnt = 64;
            for (int j = 0; j < cnt; ++j) {
                const int n = base + j;
                sum = fmaf(sa[n],  cs, sum);
                sum = fmaf(-sb[n], sn, sum);
                const float cs2 = cs * cd - sn * sd;    // theta += delta
                sn = fmaf(cs, sd, sn * cd);
                cs = cs2;
            }
        }
        out[cum + p] = sum;                             // every pixel written once
    }
}

// ---------------------------------------------------------------------------
extern "C" void kernel_launch(void* const* d_in, const int* in_sizes, int n_in,
                              void* d_out, int out_size, void* d_ws, size_t ws_size,
                              hipStream_t stream) {
    const float* x_re = (const float*)d_in[0];   // (LMAX, MMAX) f32
    const float* x_im = (const float*)d_in[1];   // (LMAX, MMAX) f32
    const float* pct  = (const float*)d_in[2];   // (MMAX, LMAX, LMAX) f32
    float* out = (float*)d_out;                  // NPIX f32
    float* ftm = (float*)d_ws;                   // (LMAX, MMAX, 2) f32 ~ 1.05 MB

    // Stage 1: 257 m-columns x 4 k-tile groups; 8 waves/block, 1 tile/wave.
    dim3 gA(MMAXD, 4, 1);
    legendre_wmma_kernel<<<gA, 256, 0, stream>>>(x_re, x_im, pct, ftm);

    // Stage 2: one block per ring.
    ring_synth_kernel<<<LMAXD, 256, 0, stream>>>(ftm, out);
}